// GCNEncoder_9491877724922
// MI455X (gfx1250) — compile-verified
//
#include <hip/hip_runtime.h>

// ---------------------------------------------------------------------------
// GCN encoder for MI455X (gfx1250).  fp32 end-to-end; GEMMs use
// V_WMMA_F32_16X16X4_F32 (full-precision CDNA5 matrix op).
//
// GEMM inner loop uses a "fragment-ready" LDS layout so each lane fetches its
// whole B fragment with one aligned ds_load_b64 (conflict-free: 32 lanes x
// 2 dwords = 64 banks) straight into an even-aligned VGPR pair.
// ---------------------------------------------------------------------------

typedef __attribute__((ext_vector_type(2))) float v2f;
typedef __attribute__((ext_vector_type(8))) float v8f;

static inline int cdiv(long long a, long long b) { return (int)((a + b - 1) / b); }

// ---------------------------------------------------------------------------
// Tiled GEMM: H[nrows,N] = X[nrows,K] @ W^T, W given row-major [N,K].
// One block = 8 waves, each wave computes a 16-row strip x all N columns.
//
// A-fragment layout per CDNA5 ISA (32-bit A 16x4):
//   lane<16 -> K = k0 + c,  lane>=16 -> K = k0 + 2 + c   (c = vector component)
// B-fragment mirrors the C/D layout: component r, lane-group g -> K = k0+r+2g.
//
// LDS fragment layout: BF[((k0/4)*NT + t)*64 + lane*2 + r]
//   holds W^T[k0 + r + 2*(lane>>4)][t*16 + (lane&15)] = W[n][k].
// ---------------------------------------------------------------------------
template <int K, int N>
__launch_bounds__(256)
__global__ void gemm_wmma_f32(const float* __restrict__ X,
                              const float* __restrict__ W,   // [N][K] row-major
                              float* __restrict__ H,         // [nrows][N]
                              int nrows) {
    constexpr int NT = N / 16;
    __shared__ float BF[(K / 4) * NT * 64];                   // == K*N floats

    const int tid = threadIdx.x;
    // Coalesced read of W; scatter into fragment-ready LDS layout.
    for (int idx = tid; idx < K * N; idx += 256) {
        const int n  = idx / K;
        const int k  = idx - n * K;
        const int ks = k >> 2;            // k-step index
        const int kk = k & 3;             // offset within step
        const int g  = kk >> 1;           // lane group (0: lanes 0-15, 1: 16-31)
        const int r  = kk & 1;            // fragment component (b.x / b.y)
        const int t  = n >> 4;            // n-tile
        const int nl = n & 15;            // lane-low
        BF[((ks * NT + t) * 32 + g * 16 + nl) * 2 + r] = W[idx];
    }
    __syncthreads();

    const int wave   = tid >> 5;
    const int lane   = tid & 31;
    const int laneLo = lane & 15;
    const int laneHi = lane >> 4;                             // 0 or 1
    const int m0     = (blockIdx.x * 8 + wave) * 16;
    if (m0 >= nrows) return;                                  // wave-uniform (nrows % 16 == 0)

    v8f acc[NT] = {};

    const float* xrow  = X + (size_t)(m0 + laneLo) * K;
    const float* fbase = BF + lane * 2;
    for (int ks = 0; ks < K / 4; ++ks) {
        const v2f a = *(const v2f*)(xrow + ks * 4 + 2 * laneHi);   // K = 4ks+2g+c
        const float* fb = fbase + (ks * NT) * 64;
#pragma unroll
        for (int t = 0; t < NT; ++t) {
            const v2f b = *(const v2f*)(fb + t * 64);              // one ds_load_b64
            acc[t] = __builtin_amdgcn_wmma_f32_16x16x4_f32(
                false, a, false, b, (short)0, acc[t], false, false);
        }
    }

    // C/D layout: VGPR r, lane<16 -> M = r, lane>=16 -> M = r+8; N = laneLo.
#pragma unroll
    for (int t = 0; t < NT; ++t)
#pragma unroll
        for (int r = 0; r < 8; ++r)
            H[(size_t)(m0 + r + 8 * laneHi) * N + t * 16 + laneLo] = acc[t][r];
}

// ---------------------------------------------------------------------------
// Edge-phase kernels
// ---------------------------------------------------------------------------
__global__ void k_fill1(float* __restrict__ deg, int n) {
    int i = blockIdx.x * blockDim.x + threadIdx.x;
    if (i < n) deg[i] = 1.0f;                                 // self-loop
}

__global__ void k_deg(const int* __restrict__ dst, float* __restrict__ deg, int E) {
    int e = blockIdx.x * blockDim.x + threadIdx.x;
    if (e < E) atomicAdd(&deg[dst[e]], 1.0f);                 // exact: small int counts
}

__global__ void k_rsqrt(float* __restrict__ d, int n) {
    int i = blockIdx.x * blockDim.x + threadIdx.x;
    if (i < n) d[i] = rsqrtf(d[i]);                           // deg >= 1 always
}

__global__ void k_norm(const int* __restrict__ src, const int* __restrict__ dst,
                       const float* __restrict__ dinv, float* __restrict__ norm, int E) {
    int e = blockIdx.x * blockDim.x + threadIdx.x;
    if (e < E) norm[e] = dinv[src[e]] * dinv[dst[e]];
}

// out[i][:] = H[i][:] * dinv[i]^2 + bias[:]   (self-loop message + bias)
template <int C>
__global__ void k_selfinit(const float* __restrict__ H, const float* __restrict__ dinv,
                           const float* __restrict__ bias, float* __restrict__ out, int n) {
    int gid = blockIdx.x * blockDim.x + threadIdx.x;          // over n * C/4
    const int CH = C / 4;
    int i = gid / CH;
    if (i >= n) return;
    int c = (gid - i * CH) * 4;
    float s = dinv[i] * dinv[i];
    float4 h = *(const float4*)(H + (size_t)i * C + c);
    float4 o;
    o.x = h.x * s + bias[c + 0];
    o.y = h.y * s + bias[c + 1];
    o.z = h.z * s + bias[c + 2];
    o.w = h.w * s + bias[c + 3];
    *(float4*)(out + (size_t)i * C + c) = o;
}

// out[dst[e]][:] += H[src[e]][:] * norm[e]   (L2-resident gather + atomics)
template <int C>
__global__ void k_scatter(const int* __restrict__ src, const int* __restrict__ dst,
                          const float* __restrict__ norm, const float* __restrict__ H,
                          float* __restrict__ out, int E) {
    int gid = blockIdx.x * blockDim.x + threadIdx.x;          // over E * C/4
    const int CH = C / 4;
    int e = gid / CH;
    if (e >= E) return;
    int c = (gid - e * CH) * 4;
    float nrm = norm[e];
    float4 h = *(const float4*)(H + (size_t)src[e] * C + c);
    float* o = out + (size_t)dst[e] * C + c;
    atomicAdd(o + 0, h.x * nrm);
    atomicAdd(o + 1, h.y * nrm);
    atomicAdd(o + 2, h.z * nrm);
    atomicAdd(o + 3, h.w * nrm);
}

__global__ void k_relu(const float* __restrict__ in, float* __restrict__ out, int n) {
    int i = blockIdx.x * blockDim.x + threadIdx.x;
    if (i < n) out[i] = fmaxf(in[i], 0.0f);
}

// ---------------------------------------------------------------------------
// Host launcher
// ---------------------------------------------------------------------------
extern "C" void kernel_launch(void* const* d_in, const int* in_sizes, int n_in,
                              void* d_out, int out_size, void* d_ws, size_t ws_size,
                              hipStream_t stream) {
    constexpr int IN  = 256;
    constexpr int HID = 128;
    constexpr int LAT = 64;

    const float* x   = (const float*)d_in[0];
    const int*   ei  = (const int*)d_in[1];   // [2, E] int32: row0 = src, row1 = dst
    const float* W1  = (const float*)d_in[2]; // [HID, IN]
    const float* b1  = (const float*)d_in[3]; // [HID]
    const float* W2  = (const float*)d_in[4]; // [LAT, HID]
    const float* b2  = (const float*)d_in[5]; // [LAT]
    float*       out = (float*)d_out;         // [N, LAT]

    const int N = in_sizes[0] / IN;
    const int E = in_sizes[1] / 2;
    const int* src = ei;
    const int* dst = ei + E;

    // Workspace carve-up (all chunks stay 16B aligned).
    char* ws = (char*)d_ws;
    float* dinv = (float*)ws;  ws += (size_t)N * sizeof(float);
    float* norm = (float*)ws;  ws += (size_t)E * sizeof(float);
    float* H1   = (float*)ws;  ws += (size_t)N * HID * sizeof(float);  // 51.2 MB
    float* S1   = (float*)ws;                                          // 51.2 MB (also reused as H2)
    float* H2   = S1;

    const int T = 256;

    // 1) degrees (incl. self-loop) -> dinv, edge norms
    k_fill1<<<cdiv(N, T), T, 0, stream>>>(dinv, N);
    k_deg  <<<cdiv(E, T), T, 0, stream>>>(dst, dinv, E);
    k_rsqrt<<<cdiv(N, T), T, 0, stream>>>(dinv, N);
    k_norm <<<cdiv(E, T), T, 0, stream>>>(src, dst, dinv, norm, E);

    // 2) layer 1: H1 = X @ W1^T  (fp32 WMMA)
    gemm_wmma_f32<IN, HID><<<cdiv(N, 128), T, 0, stream>>>(x, W1, H1, N);

    // 3) layer 1 aggregate: S1 = selfloop+bias, += edge messages, then ReLU -> H1
    k_selfinit<HID><<<cdiv((long long)N * (HID / 4), T), T, 0, stream>>>(H1, dinv, b1, S1, N);
    k_scatter<HID><<<cdiv((long long)E * (HID / 4), T), T, 0, stream>>>(src, dst, norm, H1, S1, E);
    k_relu<<<cdiv((long long)N * HID, T), T, 0, stream>>>(S1, H1, N * HID);

    // 4) layer 2: H2 = relu_h @ W2^T  (fp32 WMMA), H2 aliases S1
    gemm_wmma_f32<HID, LAT><<<cdiv(N, 128), T, 0, stream>>>(H1, W2, H2, N);

    // 5) layer 2 aggregate directly into d_out
    k_selfinit<LAT><<<cdiv((long long)N * (LAT / 4), T), T, 0, stream>>>(H2, dinv, b2, out, N);
    k_scatter<LAT><<<cdiv((long long)E * (LAT / 4), T), T, 0, stream>>>(src, dst, norm, H2, out, E);
}